// Complex_Linear_DS_agent_20839181320485
// MI455X (gfx1250) — compile-verified
//
#include <hip/hip_runtime.h>
#include <hip/hip_bf16.h>
#include <math.h>

typedef __attribute__((ext_vector_type(2))) float v2f;
typedef __attribute__((ext_vector_type(8))) float v8f;
typedef __attribute__((ext_vector_type(4))) int   v4i;

#define HS 256
#define NT 1024
#define NSTEPS 800
#define CHUNK 32
#define NCHUNK 25          // 25 * 32 = 800
#define COLS 32            // columns of x per workgroup
#define XPITCH 40          // x-tile pitch: 2*40 % 64 == 16 -> half-waves hit disjoint banks
#define MPITCH 260         // M pitch: row step = 4 banks; +2 offset of hi half -> disjoint b64 bank pairs
#define ACT_ELEMS (NT * 2 * NSTEPS)

#if __has_builtin(__builtin_amdgcn_global_load_async_to_lds_b128)
#define ASYNC_LDS 1
#endif

// ---------------------------------------------------------------------------
// Phase A: Mm = I - 0.01 * A   (Euler step matrix)
// ---------------------------------------------------------------------------
__global__ void build_mm_kernel(const float* __restrict__ A, float* __restrict__ Mm) {
  int idx = blockIdx.x * 256 + threadIdx.x;     // 256 blocks x 256 threads = 65536
  int i = idx >> 8, j = idx & 255;
  float v = -0.01f * A[idx];
  if (i == j) v += 1.0f;
  Mm[idx] = v;
}

// ---------------------------------------------------------------------------
// Phase A2: h[n][j] = relu(target[n,:] @ W1[j,:] + b1[j])   (K=2, scalar is fine)
// ---------------------------------------------------------------------------
__global__ void init_h_kernel(const float* __restrict__ target,
                              const float* __restrict__ W1,
                              const float* __restrict__ b1,
                              float* __restrict__ h) {
  int n = blockIdx.x;          // 1024
  int j = threadIdx.x;         // 256
  float t0 = target[n * 2 + 0], t1 = target[n * 2 + 1];
  float s = t0 * W1[j * 2 + 0] + t1 * W1[j * 2 + 1] + b1[j];
  h[n * 256 + j] = s > 0.0f ? s : 0.0f;
}

// ---------------------------------------------------------------------------
// Generic fp32 WMMA GEMM: C(256 x N) = A(256x256) @ B(256xN) [+ bias per row]
// BT=1: B stored as (N x 256) row-major (logical B[k][n] = B[n*256+k]).
// Compile-time BT/BIAS -> branch-free inner loop.
// ---------------------------------------------------------------------------
template <int BT, int BIAS>
__global__ void __launch_bounds__(256)
gemm256_wmma_kernel(const float* __restrict__ A, const float* __restrict__ B,
                    float* __restrict__ C, int ldb, int ldc,
                    const float* __restrict__ bias) {
  const int lane = threadIdx.x & 31;
  const int w    = threadIdx.x >> 5;
  const int hl   = lane >> 4;           // half-wave select (k offset)
  const int l16  = lane & 15;
  const int rt   = blockIdx.y * 8 + w;  // 0..15
  const int ct   = blockIdx.x;          // 0..N/16-1

  v8f acc = {};
  const float* arow = A + (rt * 16 + l16) * 256;
#pragma unroll 4
  for (int kk = 0; kk < 64; ++kk) {
    const int k0 = kk * 4 + 2 * hl;
    v2f a = *(const v2f*)(arow + k0);            // 8B-aligned (k0 even)
    v2f b;
    if (BT) {
      b = *(const v2f*)(B + (ct * 16 + l16) * 256 + k0);
    } else {
      b.x = B[(k0 + 0) * ldb + ct * 16 + l16];
      b.y = B[(k0 + 1) * ldb + ct * 16 + l16];
    }
    acc = __builtin_amdgcn_wmma_f32_16x16x4_f32(false, a, false, b,
                                                (short)0, acc, false, false);
  }
#pragma unroll
  for (int v = 0; v < 8; ++v) {
    const int m = rt * 16 + v + 8 * hl;
    float cv = acc[v];
    if (BIAS) cv += bias[m];
    C[(size_t)m * ldc + ct * 16 + l16] = cv;
  }
}

// ---------------------------------------------------------------------------
// Phase D: per (chunk, column-tile) run 32 local Euler steps.
// Entire M (256x256 fp32) staged once into LDS (266 KB, pitch 260) via async
// global->LDS b128 copies; x tile (256x32) also in LDS. 309 KB total fits the
// 320 KB WGP LDS. Each wave owns 4 output tiles, accumulators in VGPRs.
// ---------------------------------------------------------------------------
__global__ void __launch_bounds__(256)
recur_kernel(const float* __restrict__ Mm, const float* __restrict__ Wm,
             const float* __restrict__ bm, float* __restrict__ out) {
  __shared__ float mS[256 * MPITCH];   // 266,240 B : full Euler matrix
  __shared__ float xs[256 * XPITCH];   //  40,960 B : x tile
  __shared__ float wmS[2 * 256];       //   2,048 B : Wm

  const int t     = threadIdx.x;
  const int ctile = blockIdx.x;        // 0..31
  const int c     = blockIdx.y;        // 0..24
  const int n0    = ctile * COLS;
  float* hid = out + ACT_ELEMS;
  const float* src = hid + (size_t)(c * CHUNK) * (HS * NT);  // chunk start x_{32c}

  wmS[t]       = Wm[t];
  wmS[256 + t] = Wm[256 + t];

  // ---- stage M into LDS: thread t covers cols (t&63)*4..+3, rows (t>>6)+4i
  {
    const int cg = (t & 63) * 4;
    const int rg = t >> 6;
#if ASYNC_LDS
    for (int i = 0; i < 64; ++i) {
      const int r = rg + 4 * i;
      __builtin_amdgcn_global_load_async_to_lds_b128(
          (v4i*)(Mm + r * 256 + cg), (v4i*)&mS[r * MPITCH + cg], 0, 0);
    }
#if __has_builtin(__builtin_amdgcn_s_wait_asynccnt)
    __builtin_amdgcn_s_wait_asynccnt(0);
#else
    asm volatile("s_wait_asynccnt 0x0" ::: "memory");
#endif
#else
    for (int i = 0; i < 64; ++i) {
      const int r = rg + 4 * i;
      *(float4*)&mS[r * MPITCH + cg] = *(const float4*)(Mm + r * 256 + cg);
    }
#endif
  }
  // ---- stage chunk-start x tile (b128 both sides)
  {
    const int c4 = (t & 7) * 4;
    const int r0 = t >> 3;
    for (int i = 0; i < 8; ++i) {
      const int r = r0 + 32 * i;
      *(float4*)&xs[r * XPITCH + c4] = *(const float4*)(src + (size_t)r * NT + n0 + c4);
    }
  }
  __syncthreads();

  const int lane = t & 31, w = t >> 5;
  const int hl = lane >> 4, l16 = lane & 15;
  const int rtA = w * 2, rtB = w * 2 + 1;
  const int offA = (rtA * 16 + l16) * MPITCH;
  const int offB = (rtB * 16 + l16) * MPITCH;
  const float bm0 = bm[0], bm1 = bm[1];

  for (int j = 0; j < CHUNK; ++j) {
    const int step = c * CHUNK + j;
    float* hidT = hid + (size_t)step * (HS * NT);

    // 1) store x_t (pre-update) to hidden_states, 128B-coalesced b128 stores
    {
      const int c4 = (t & 7) * 4;
      const int r0 = t >> 3;
      for (int i = 0; i < 8; ++i) {
        const int r = r0 + 32 * i;
        *(float4*)(hidT + (size_t)r * NT + n0 + c4) = *(const float4*)&xs[r * XPITCH + c4];
      }
    }
    // 2) actions[n][a][step] = tanh(Wm[a,:] @ x[:,n] + bm[a])
    if (t < 64) {
      const int a = t >> 5, nl = t & 31;
      float s = (a == 0) ? bm0 : bm1;
      for (int hh = 0; hh < 256; ++hh)
        s += wmS[a * 256 + hh] * xs[hh * XPITCH + nl];
      out[(size_t)(n0 + nl) * (2 * NSTEPS) + a * NSTEPS + step] = tanhf(s);
    }
    // 3) x_{t+1} = M @ x_t : all operands from LDS, fp32 WMMA
    v8f acc00 = {}, acc01 = {}, acc10 = {}, acc11 = {};
#pragma unroll 4
    for (int kk = 0; kk < 64; ++kk) {
      const int k0 = kk * 4 + 2 * hl;
      v2f aA = *(const v2f*)(&mS[offA + k0]);     // ds_load_b64, bank-disjoint halves
      v2f aB = *(const v2f*)(&mS[offB + k0]);
      v2f b0, b1;
      b0.x = xs[(k0 + 0) * XPITCH + l16];
      b0.y = xs[(k0 + 1) * XPITCH + l16];
      b1.x = xs[(k0 + 0) * XPITCH + 16 + l16];
      b1.y = xs[(k0 + 1) * XPITCH + 16 + l16];
      acc00 = __builtin_amdgcn_wmma_f32_16x16x4_f32(false, aA, false, b0, (short)0, acc00, false, false);
      acc01 = __builtin_amdgcn_wmma_f32_16x16x4_f32(false, aA, false, b1, (short)0, acc01, false, false);
      acc10 = __builtin_amdgcn_wmma_f32_16x16x4_f32(false, aB, false, b0, (short)0, acc10, false, false);
      acc11 = __builtin_amdgcn_wmma_f32_16x16x4_f32(false, aB, false, b1, (short)0, acc11, false, false);
    }
    __syncthreads();               // all reads of xs done
#pragma unroll
    for (int v = 0; v < 8; ++v) {
      const int mA = rtA * 16 + v + 8 * hl;
      const int mB = rtB * 16 + v + 8 * hl;
      xs[mA * XPITCH + l16]      = acc00[v];
      xs[mA * XPITCH + 16 + l16] = acc01[v];
      xs[mB * XPITCH + l16]      = acc10[v];
      xs[mB * XPITCH + 16 + l16] = acc11[v];
    }
    __syncthreads();               // xs now holds x_{t+1}
  }
}

// ---------------------------------------------------------------------------
extern "C" void kernel_launch(void* const* d_in, const int* in_sizes, int n_in,
                              void* d_out, int out_size, void* d_ws, size_t ws_size,
                              hipStream_t stream) {
  (void)in_sizes; (void)n_in; (void)out_size; (void)ws_size;
  const float* target = (const float*)d_in[0];
  const float* A      = (const float*)d_in[1];
  const float* W1     = (const float*)d_in[2];
  const float* b1     = (const float*)d_in[3];
  const float* W2     = (const float*)d_in[4];
  const float* b2     = (const float*)d_in[5];
  const float* Wm     = (const float*)d_in[6];
  const float* bm     = (const float*)d_in[7];
  float* out = (float*)d_out;

  float* ws   = (float*)d_ws;
  float* Mm   = ws;                 // 256*256
  float* bufA = ws + 65536;         // 256*256
  float* bufB = ws + 131072;        // 256*256
  float* h    = ws + 196608;        // 1024*256
  float* hid  = out + ACT_ELEMS;    // hidden_states region of d_out

  // Euler matrix and MLP hidden layer
  build_mm_kernel<<<256, 256, 0, stream>>>(A, Mm);
  init_h_kernel<<<1024, 256, 0, stream>>>(target, W1, b1, h);

  // x0 = W2 @ h^T + b2  -> written directly into hidden_states[0]
  gemm256_wmma_kernel<1, 1><<<dim3(64, 2), 256, 0, stream>>>(W2, h, hid, 256, 1024, b2);

  // S = Mm^32 by 5 repeated squarings (ends in bufA)
  gemm256_wmma_kernel<0, 0><<<dim3(16, 2), 256, 0, stream>>>(Mm,   Mm,   bufA, 256, 256, nullptr);
  gemm256_wmma_kernel<0, 0><<<dim3(16, 2), 256, 0, stream>>>(bufA, bufA, bufB, 256, 256, nullptr);
  gemm256_wmma_kernel<0, 0><<<dim3(16, 2), 256, 0, stream>>>(bufB, bufB, bufA, 256, 256, nullptr);
  gemm256_wmma_kernel<0, 0><<<dim3(16, 2), 256, 0, stream>>>(bufA, bufA, bufB, 256, 256, nullptr);
  gemm256_wmma_kernel<0, 0><<<dim3(16, 2), 256, 0, stream>>>(bufB, bufB, bufA, 256, 256, nullptr);
  float* S = bufA;

  // chunk starts: x_{32c} = S @ x_{32(c-1)}, written into their final
  // hidden_states slots (they ARE hidden_states[32c])
  for (int cc = 1; cc < NCHUNK; ++cc) {
    gemm256_wmma_kernel<0, 0><<<dim3(64, 2), 256, 0, stream>>>(
        S,
        hid + (size_t)(cc - 1) * CHUNK * HS * NT,
        hid + (size_t)cc * CHUNK * HS * NT,
        1024, 1024, nullptr);
  }

  // 25 chunks x 32 column-tiles = 800 workgroups, 32 steps each
  recur_kernel<<<dim3(32, NCHUNK), 256, 0, stream>>>(Mm, Wm, bm, out);
}